// ContrastCELoss_4458176053806
// MI455X (gfx1250) — compile-verified
//
#include <hip/hip_runtime.h>
#include <stdint.h>

// ---------------------------------------------------------------------------
// ContrastCELoss for MI455X (gfx1250).
// Heavy part: G = F F^T, F=[8192,256] fp32  -> v_wmma_f32_16x16x4_f32.
// A-block AND streaming B tiles staged to LDS via the Tensor Data Mover
// (tensor_load_to_lds, double-buffered, s_wait_tensorcnt), descriptor
// pad_enable giving a bank-conflict-free 260-float LDS row stride.
// ---------------------------------------------------------------------------

typedef __attribute__((ext_vector_type(2))) float v2f;
typedef __attribute__((ext_vector_type(8))) float v8f;
typedef __attribute__((ext_vector_type(4))) unsigned int v4u;
typedef __attribute__((ext_vector_type(4))) int v4i;
typedef __attribute__((ext_vector_type(8))) int v8i;

#define NB 8
#define NC 21
#define HW 16384
#define ND 256
#define NROW 8192
#define NHALF 4096
#define INV_TEMP (1.0f/0.07f)
#define NEG_BIG (-3.402823466e38f)
#define LDA 260   // 256 + 4 pad floats per row

// workspace layout (float units)
#define OFF_F    0ull
#define OFF_SEL  (OFF_F + (size_t)NROW*ND)     // 8192 ints
#define OFF_PM   (OFF_SEL + (size_t)NROW)
#define OFF_PS   (OFF_PM + 8ull*NROW)
#define OFF_PP   (OFF_PS + 8ull*NROW)
#define OFF_PQ   (OFF_PP + 8ull*NROW)
#define OFF_RM   (OFF_PQ + 8ull*NROW)
#define OFF_RS   (OFF_RM + (size_t)NROW)
#define OFF_RP   (OFF_RS + (size_t)NROW)
#define OFF_ACC  (OFF_RP + (size_t)NROW)

#if defined(__HIP_DEVICE_COMPILE__) && defined(__gfx1250__) && \
    __has_builtin(__builtin_amdgcn_tensor_load_to_lds) && \
    __has_builtin(__builtin_amdgcn_s_wait_tensorcnt)
#define HAVE_TDM 1
#else
#define HAVE_TDM 0
#endif

#if defined(__HIP_DEVICE_COMPILE__) && defined(__gfx1250__)
#if HAVE_TDM
#warning "CDNA5 probe: TDM tensor_load_to_lds path ENABLED"
#else
#warning "CDNA5 probe: TDM builtins NOT available -> cooperative-load fallback"
#endif
#endif

#if HAVE_TDM
// Build D# per CDNA5 ISA and issue TENSOR_LOAD_TO_LDS.
// 2D tile: 256 elems/row (4B), `rows` rows, tensor row stride 256 elems,
// LDS padding 4 dwords after every 256 dwords -> LDS row stride = 260 floats.
__device__ __forceinline__ void tdm_load_rows(const float* gsrc, unsigned lds_byte, int rows) {
  unsigned long long ga = (unsigned long long)(size_t)gsrc;
  v4u g0;
  g0.x = 1u;                                               // count=1 (valid user D#)
  g0.y = lds_byte;                                         // lds_addr
  g0.z = (unsigned)ga;                                     // global_addr[31:0]
  g0.w = (unsigned)((ga >> 32) & 0x1FFFFFFull) | (2u<<30); // addr[56:32] | type=2
  v8i g1;
  g1.s0 = (int)((2u<<16) | (1u<<20) | (7u<<22) | (3u<<25)); // 4B elems, pad en, 256dw interval, 4dw pad
  g1.s1 = (int)(256u << 16);    // tensor_dim0 = 256 (bits 63:48)
  g1.s2 = (int)(8192u << 16);   // tensor_dim1 = 8192 (bits 95:80)
  g1.s3 = (int)(256u << 16);    // tile_dim0 = 256 (bits 127:112)
  g1.s4 = rows & 0xFFFF;        // tile_dim1
  g1.s5 = 256;                  // tensor_dim0_stride (low 32 of 48)
  g1.s6 = 0;
  g1.s7 = 0;
  v4i z4 = {0,0,0,0};
#if __clang_major__ >= 23
  v8i z8 = {0,0,0,0,0,0,0,0};
  __builtin_amdgcn_tensor_load_to_lds(g0, g1, z4, z4, z8, 0);
#else
  __builtin_amdgcn_tensor_load_to_lds(g0, g1, z4, z4, 0);
#endif
}
#endif

// Load 128 rows x 256 cols of F into LDS (row stride LDA), then barrier.
__device__ __forceinline__ void load_Ablock(const float* __restrict__ F, float* As,
                                            int m0, int tid) {
#if HAVE_TDM
  if (tid < 32) {   // wave 0 issues the DMA; EXEC ignored by TDM
    tdm_load_rows(&F[(size_t)m0*ND], (unsigned)(size_t)As, 128);
    __builtin_amdgcn_s_wait_tensorcnt(0);
  }
#else
#pragma unroll
  for (int rr = 0; rr < 8; ++rr) {
    int row = rr*16 + (tid >> 4);
    int cb  = (tid & 15) * 16;
    const float4* s = (const float4*)&F[(size_t)(m0 + row)*ND + cb];
    float4* d = (float4*)&As[row*LDA + cb];
    d[0]=s[0]; d[1]=s[1]; d[2]=s[2]; d[3]=s[3];
  }
#endif
  __syncthreads();
}

// Fallback cooperative 16x256 B-tile load (only compiled when !HAVE_TDM).
__device__ __forceinline__ void coop_load_Btile(const float* __restrict__ F, float* Bs,
                                                int n0, int tid) {
  const int r  = tid >> 4;
  const int cb = (tid & 15) * 16;
  const float4* s = (const float4*)&F[(size_t)(n0 + r)*ND + cb];
  float4* d = (float4*)&Bs[r*LDA + cb];
  d[0]=s[0]; d[1]=s[1]; d[2]=s[2]; d[3]=s[3];
}

// ---------------------------------------------------------------------------
// 0) init: zero pass-B partials and CE accumulators
__global__ void cl_init_kernel(float* __restrict__ pq, float* __restrict__ acc) {
  int i = blockIdx.x*256 + threadIdx.x;
  if (i < 8*NROW) pq[i] = 0.f;
  if (i < 8) acc[i] = 0.f;
}

// ---------------------------------------------------------------------------
// 1) weighted cross-entropy: one thread per pixel, classes strided -> coalesced
__global__ __launch_bounds__(256) void cl_ce_kernel(const float* __restrict__ seg,
                                                    const int* __restrict__ tgt,
                                                    const float* __restrict__ cw,
                                                    float* __restrict__ acc) {
  int p = blockIdx.x*256 + threadIdx.x;      // 0..131071 over (b, pix)
  int b = p >> 14, pix = p & (HW-1);
  const float* s = seg + ((size_t)b*NC)*HW + pix;
  float x[NC];
  float mx = NEG_BIG;
#pragma unroll
  for (int c = 0; c < NC; ++c) { x[c] = s[(size_t)c*HW]; mx = fmaxf(mx, x[c]); }
  float sum = 0.f;
#pragma unroll
  for (int c = 0; c < NC; ++c) sum += __expf(x[c] - mx);
  float lse = __logf(sum) + mx;
  int t = tgt[p];
  bool valid = (t != 255);
  int st = valid ? t : 0;
  float w   = valid ? cw[st] : 0.f;
  float wnl = valid ? w * (lse - x[st]) : 0.f;
  __shared__ float r1[256], r2[256];
  r1[threadIdx.x] = wnl; r2[threadIdx.x] = w;
  __syncthreads();
  for (int o = 128; o > 0; o >>= 1) {
    if (threadIdx.x < o) { r1[threadIdx.x] += r1[threadIdx.x+o]; r2[threadIdx.x] += r2[threadIdx.x+o]; }
    __syncthreads();
  }
  if (threadIdx.x == 0) { atomicAdd(&acc[0], r1[0]); atomicAdd(&acc[1], r2[0]); }
}

// ---------------------------------------------------------------------------
// 2) per-batch bitonic sort of cam -> top-512 (fg) and bottom-512 (bg) pixels.
//    Loss is permutation-invariant within each label group, only the SET matters.
__global__ __launch_bounds__(1024) void cl_topk_kernel(const float* __restrict__ cam,
                                                       int* __restrict__ sel) {
  __shared__ float sv[HW];
  __shared__ int   si[HW];
  int b = blockIdx.x;
  const float* c = cam + (size_t)b*HW;
  for (int i = threadIdx.x; i < HW; i += 1024) { sv[i] = c[i]; si[i] = i; }
  __syncthreads();
  for (int k = 2; k <= HW; k <<= 1) {
    for (int j = k >> 1; j > 0; j >>= 1) {
      for (int i = threadIdx.x; i < HW; i += 1024) {
        int ij = i ^ j;
        if (ij > i) {
          bool desc = ((i & k) == 0);            // overall descending
          bool sw = desc ? (sv[i] < sv[ij]) : (sv[i] > sv[ij]);
          if (sw) {
            float tv = sv[i]; sv[i] = sv[ij]; sv[ij] = tv;
            int   ti = si[i]; si[i] = si[ij]; si[ij] = ti;
          }
        }
      }
      __syncthreads();
    }
  }
  for (int s = threadIdx.x; s < 512; s += 1024) {
    sel[b*512 + s]          = si[s];            // fg: highest activations
    sel[NHALF + b*512 + s]  = si[HW - 1 - s];   // bg: lowest activations
  }
}

// ---------------------------------------------------------------------------
// 3) gather selected pixel features into F[8192][256] row-major
__global__ __launch_bounds__(256) void cl_gather_kernel(const float* __restrict__ embed,
                                                        const int* __restrict__ sel,
                                                        float* __restrict__ F) {
  int row = blockIdx.x, d = threadIdx.x;
  int b = (row < NHALF) ? (row >> 9) : ((row - NHALF) >> 9);
  int pix = sel[row];
  F[(size_t)row*ND + d] = embed[((size_t)b*ND + d)*HW + pix];
}

// ---------------------------------------------------------------------------
// 4) Pass A: per (128-row block, 1024-col stripe): flash running max,
//    neg_sum (opposite-label), pos logit sum -> striped partials.
__global__ __launch_bounds__(256) void cl_passA_kernel(const float* __restrict__ F,
    float* __restrict__ pm, float* __restrict__ ps, float* __restrict__ pp) {
  __shared__ float As[128*LDA];
  __shared__ float Bs[2][16*LDA];
  const int wave = threadIdx.x >> 5;
  const int lane = threadIdx.x & 31;
  const int m0 = blockIdx.x * 128;
  const int stripe = blockIdx.y;

  load_Ablock(F, As, m0, threadIdx.x);

  const int hb = lane >> 4;            // half-wave
  const int kb = hb * 2;               // K offset of this half
  const int arow = wave*16 + (lane & 15);
  v2f areg[64];                        // full K=256 A fragments for our 16 rows
#pragma unroll
  for (int kk = 0; kk < 64; ++kk)
    areg[kk] = *(const v2f*)&As[arow*LDA + kk*4 + kb];

  float mrun[8], srun[8], prun[8];
#pragma unroll
  for (int r = 0; r < 8; ++r) { mrun[r] = NEG_BIG; srun[r] = 0.f; prun[r] = 0.f; }

  const bool rowFg = (m0 < NHALF);
  const int jcol = lane & 15;
  const int brow = jcol * LDA;

#if HAVE_TDM
  if (threadIdx.x < 32)   // prologue: DMA B tile 0 into ping buffer
    tdm_load_rows(&F[(size_t)(stripe*1024)*ND], (unsigned)(size_t)&Bs[0][0], 16);
#endif

  for (int t = 0; t < 64; ++t) {
    const int n0 = (stripe*64 + t) * 16;
#if HAVE_TDM
    if (threadIdx.x < 32) __builtin_amdgcn_s_wait_tensorcnt(0);  // tile t ready
    __syncthreads();                                             // publish; prev compute done
    if (t + 1 < 64 && threadIdx.x < 32)                          // overlap DMA of t+1
      tdm_load_rows(&F[(size_t)(n0 + 16)*ND], (unsigned)(size_t)&Bs[(t+1)&1][0], 16);
    const float* Bt = &Bs[t & 1][0];
#else
    __syncthreads();
    coop_load_Btile(F, &Bs[0][0], n0, threadIdx.x);
    __syncthreads();
    const float* Bt = &Bs[0][0];
#endif

    v8f c = {};
#pragma unroll
    for (int kk = 0; kk < 64; ++kk) {
      v2f bv = *(const v2f*)&Bt[brow + kk*4 + kb];
      c = __builtin_amdgcn_wmma_f32_16x16x4_f32(false, areg[kk], false, bv,
                                                (short)0, c, false, false);
    }

    const bool same = (rowFg == (n0 < NHALF));
    const int jg = n0 + jcol;
#pragma unroll
    for (int r = 0; r < 8; ++r) {
      const float l = c[r] * INV_TEMP;
      const float mn = fmaxf(mrun[r], l);
      srun[r] = srun[r]*__expf(mrun[r] - mn) + (same ? 0.f : __expf(l - mn));
      mrun[r] = mn;
      const int ig = m0 + wave*16 + r + 8*hb;
      if (same && jg != ig) prun[r] += l;
    }
  }

  // flash-merge across the 16 lanes holding one row
#pragma unroll
  for (int r = 0; r < 8; ++r) {
    float m = mrun[r], s = srun[r], p = prun[r];
    for (int off = 1; off < 16; off <<= 1) {
      float mo = __shfl_xor(m, off, 32);
      float so = __shfl_xor(s, off, 32);
      float po = __shfl_xor(p, off, 32);
      float M = fmaxf(m, mo);
      s = s*__expf(m - M) + so*__expf(mo - M);
      m = M; p += po;
    }
    if (jcol == 0) {
      int i = m0 + wave*16 + r + 8*hb;
      pm[(size_t)stripe*NROW + i] = m;
      ps[(size_t)stripe*NROW + i] = s;
      pp[(size_t)stripe*NROW + i] = p;
    }
  }
}

// ---------------------------------------------------------------------------
// 5) merge stripe partials -> per-row max / neg_sum / pos-logit-sum
__global__ __launch_bounds__(256) void cl_merge_kernel(const float* __restrict__ pm,
    const float* __restrict__ ps, const float* __restrict__ pp,
    float* __restrict__ rM, float* __restrict__ rS, float* __restrict__ rP) {
  int i = blockIdx.x*256 + threadIdx.x;
  float m = NEG_BIG, s = 0.f, p = 0.f;
#pragma unroll
  for (int st = 0; st < 8; ++st) {
    float mo = pm[(size_t)st*NROW + i];
    float so = ps[(size_t)st*NROW + i];
    float M = fmaxf(m, mo);
    s = s*__expf(m - M) + so*__expf(mo - M);
    m = M;
    p += pp[(size_t)st*NROW + i];
  }
  rM[i] = m; rS[i] = s; rP[i] = p;
}

// ---------------------------------------------------------------------------
// 6) Pass B: sum over pos of log(exp(l - max) + neg_sum).
//    Only same-label stripes contribute -> early uniform exit otherwise.
__global__ __launch_bounds__(256) void cl_passB_kernel(const float* __restrict__ F,
    const float* __restrict__ rM, const float* __restrict__ rS,
    float* __restrict__ pq) {
  const int m0 = blockIdx.x * 128;
  const int stripe = blockIdx.y;
  if ((m0 < NHALF) != (stripe*1024 < NHALF)) return;   // pq pre-zeroed

  __shared__ float As[128*LDA];
  __shared__ float Bs[2][16*LDA];
  const int wave = threadIdx.x >> 5;
  const int lane = threadIdx.x & 31;

  load_Ablock(F, As, m0, threadIdx.x);

  const int hb = lane >> 4;
  const int kb = hb * 2;
  const int arow = wave*16 + (lane & 15);
  v2f areg[64];
#pragma unroll
  for (int kk = 0; kk < 64; ++kk)
    areg[kk] = *(const v2f*)&As[arow*LDA + kk*4 + kb];

  const int jcol = lane & 15;
  const int brow = jcol * LDA;
  float mi[8], si[8], q[8];
#pragma unroll
  for (int r = 0; r < 8; ++r) {
    int i = m0 + wave*16 + r + 8*hb;
    mi[r] = rM[i]; si[r] = rS[i]; q[r] = 0.f;
  }

#if HAVE_TDM
  if (threadIdx.x < 32)
    tdm_load_rows(&F[(size_t)(stripe*1024)*ND], (unsigned)(size_t)&Bs[0][0], 16);
#endif

  for (int t = 0; t < 64; ++t) {
    const int n0 = (stripe*64 + t) * 16;
#if HAVE_TDM
    if (threadIdx.x < 32) __builtin_amdgcn_s_wait_tensorcnt(0);
    __syncthreads();
    if (t + 1 < 64 && threadIdx.x < 32)
      tdm_load_rows(&F[(size_t)(n0 + 16)*ND], (unsigned)(size_t)&Bs[(t+1)&1][0], 16);
    const float* Bt = &Bs[t & 1][0];
#else
    __syncthreads();
    coop_load_Btile(F, &Bs[0][0], n0, threadIdx.x);
    __syncthreads();
    const float* Bt = &Bs[0][0];
#endif

    v8f c = {};
#pragma unroll
    for (int kk = 0; kk < 64; ++kk) {
      v2f bv = *(const v2f*)&Bt[brow + kk*4 + kb];
      c = __builtin_amdgcn_wmma_f32_16x16x4_f32(false, areg[kk], false, bv,
                                                (short)0, c, false, false);
    }

    const int jg = n0 + jcol;
#pragma unroll
    for (int r = 0; r < 8; ++r) {
      const int ig = m0 + wave*16 + r + 8*hb;
      if (jg != ig) {
        float l = c[r] * INV_TEMP;
        q[r] += __logf(__expf(l - mi[r]) + si[r]);
      }
    }
  }

#pragma unroll
  for (int r = 0; r < 8; ++r) {
    float v = q[r];
    for (int off = 1; off < 16; off <<= 1) v += __shfl_xor(v, off, 32);
    if (jcol == 0)
      pq[(size_t)stripe*NROW + m0 + wave*16 + r + 8*hb] = v;
  }
}

// ---------------------------------------------------------------------------
// 7) finalize: contrast = -mean_i ((P_i - 4095*M_i - Q_i)/4095); combine w/ CE
__global__ __launch_bounds__(256) void cl_final_kernel(const float* __restrict__ rM,
    const float* __restrict__ rP, const float* __restrict__ pq,
    const float* __restrict__ acc, float* __restrict__ out) {
  __shared__ float red[256];
  float s = 0.f;
  for (int i = threadIdx.x; i < NROW; i += 256) {
    float q = 0.f;
#pragma unroll
    for (int st = 0; st < 8; ++st) q += pq[(size_t)st*NROW + i];
    s += (rP[i] - 4095.0f*rM[i] - q) * (1.0f/4095.0f);
  }
  red[threadIdx.x] = s;
  __syncthreads();
  for (int o = 128; o > 0; o >>= 1) {
    if (threadIdx.x < o) red[threadIdx.x] += red[threadIdx.x+o];
    __syncthreads();
  }
  if (threadIdx.x == 0) {
    float contrast = -(red[0] / (float)NROW);   // TEMP/BASE_TEMP == 1
    float ce = acc[0] / acc[1];
    out[0] = ce + 0.1f*contrast;
    out[1] = ce;
    out[2] = contrast;
  }
}

// ---------------------------------------------------------------------------
extern "C" void kernel_launch(void* const* d_in, const int* in_sizes, int n_in,
                              void* d_out, int out_size, void* d_ws, size_t ws_size,
                              hipStream_t stream) {
  (void)in_sizes; (void)n_in; (void)out_size; (void)ws_size;
  const float* seg    = (const float*)d_in[0];
  const float* embed  = (const float*)d_in[1];
  const int*   target = (const int*)d_in[2];
  const float* cam    = (const float*)d_in[3];
  const float* cw     = (const float*)d_in[4];
  float* out = (float*)d_out;
  float* ws  = (float*)d_ws;

  float* F   = ws + OFF_F;
  int*   sel = (int*)(ws + OFF_SEL);
  float* pm  = ws + OFF_PM;
  float* ps  = ws + OFF_PS;
  float* pp  = ws + OFF_PP;
  float* pq  = ws + OFF_PQ;
  float* rM  = ws + OFF_RM;
  float* rS  = ws + OFF_RS;
  float* rP  = ws + OFF_RP;
  float* acc = ws + OFF_ACC;

  cl_init_kernel  <<<dim3(256),  dim3(256),  0, stream>>>(pq, acc);
  cl_ce_kernel    <<<dim3(512),  dim3(256),  0, stream>>>(seg, target, cw, acc);
  cl_topk_kernel  <<<dim3(NB),   dim3(1024), 0, stream>>>(cam, sel);
  cl_gather_kernel<<<dim3(NROW), dim3(256),  0, stream>>>(embed, sel, F);
  cl_passA_kernel <<<dim3(64, 8), dim3(256), 0, stream>>>(F, pm, ps, pp);
  cl_merge_kernel <<<dim3(32),   dim3(256),  0, stream>>>(pm, ps, pp, rM, rS, rP);
  cl_passB_kernel <<<dim3(64, 8), dim3(256), 0, stream>>>(F, rM, rS, pq);
  cl_final_kernel <<<dim3(1),    dim3(256),  0, stream>>>(rM, rP, pq, acc, out);
}